// GIN_31731218383093
// MI455X (gfx1250) — compile-verified
//
#include <hip/hip_runtime.h>
#include <hip/hip_bf16.h>

typedef __attribute__((ext_vector_type(2))) float v2f;
typedef __attribute__((ext_vector_type(8))) float v8f;

#define DTOT 128
#define LDSTR 132   // LDS row stride in floats; 132 % 64 == 4 -> conflict-free WMMA A reads

// ---------------------------------------------------------------------------
// 1) concat(X, noise, h) -> x0 [N,128]
// ---------------------------------------------------------------------------
__global__ void gin_concat(const float* __restrict__ X, const float* __restrict__ noise,
                           const float* __restrict__ h, float* __restrict__ x0, int N) {
    int idx  = blockIdx.x * blockDim.x + threadIdx.x;
    int node = idx >> 7;
    int col  = idx & 127;
    if (node >= N) return;
    float v;
    if (col < 64)       v = X[node * 64 + col];
    else if (col < 96)  v = noise[node * 32 + (col - 64)];
    else                v = h[node * 32 + (col - 96)];
    x0[idx] = v;
}

// ---------------------------------------------------------------------------
// 2) zero agg (float4 granularity)
// ---------------------------------------------------------------------------
__global__ void gin_zero4(float4* __restrict__ p, int n4) {
    int idx = blockIdx.x * blockDim.x + threadIdx.x;
    if (idx < n4) p[idx] = make_float4(0.f, 0.f, 0.f, 0.f);
}

// ---------------------------------------------------------------------------
// 3) edge scatter-add: one wave32 per edge, float4 per lane (512B per edge).
//    x and agg are L2-resident (25.6MB each << 192MB L2), so this is an
//    L2 gather + global_atomic_add_f32 scatter; no HBM round trips.
// ---------------------------------------------------------------------------
__global__ void gin_scatter(const float* __restrict__ x, const int* __restrict__ src,
                            const int* __restrict__ dst, float* __restrict__ agg, int E) {
    int gid  = blockIdx.x * blockDim.x + threadIdx.x;
    int edge = gid >> 5;              // wave32: one wave per edge
    if (edge >= E) return;
    int lane = threadIdx.x & 31;
    int s = src[edge];
    int d = dst[edge];
    const float4 v = *(const float4*)(x + (size_t)s * DTOT + lane * 4);
    float* ap = agg + (size_t)d * DTOT + lane * 4;
    atomicAdd(ap + 0, v.x);
    atomicAdd(ap + 1, v.y);
    atomicAdd(ap + 2, v.z);
    atomicAdd(ap + 3, v.w);
}

// ---------------------------------------------------------------------------
// 4) y = relu(((1+eps)*x + agg) @ W + b), exact fp32 via V_WMMA_F32_16X16X4_F32.
//    Block = 8 waves; block tile = 16 rows x 128 cols; wave tile = 16x16.
//    A (16 rows x 128 K) staged in LDS once per block; W streamed from L2
//    (each element consumed exactly once per wave -> no reuse to exploit).
// ---------------------------------------------------------------------------
__global__ void __launch_bounds__(256)
gin_gemm_relu(const float* __restrict__ xin, const float* __restrict__ agg,
              const float* __restrict__ W,   const float* __restrict__ bias,
              const float* __restrict__ eps, float* __restrict__ y, int N) {
    __shared__ float Ain[16 * LDSTR];

    const int tid  = threadIdx.x;
    const int row0 = blockIdx.x * 16;
    const float s  = 1.0f + eps[0];

    // Stage (1+eps)*x + agg for this block's 16 rows: 2048 elems / 256 threads.
    #pragma unroll
    for (int i = 0; i < 8; ++i) {
        int idx  = tid + i * 256;
        int r    = idx >> 7;
        int c    = idx & 127;
        int node = row0 + r;
        float v  = 0.f;
        if (node < N) v = s * xin[(size_t)node * DTOT + c] + agg[(size_t)node * DTOT + c];
        Ain[r * LDSTR + c] = v;
    }
    __syncthreads();

    const int wave = tid >> 5;        // 0..7 -> which 16-col slab of the output
    const int lane = tid & 31;
    const int half = lane >> 4;       // 0: K,K+1   1: K+2,K+3 (A); rows 0-7 vs 8-15 (C/D)
    const int l16  = lane & 15;
    const int ncol = wave * 16 + l16;

    v8f acc = {};
    const float* wp = W + ncol;       // column ncol of row-major W[128][128]
    #pragma unroll
    for (int k0 = 0; k0 < DTOT; k0 += 4) {
        const int ka = k0 + half * 2;
        v2f a, b;
        // A fragment (16x4): lane l16 = row M, VGPR0/1 = K=ka, ka+1
        a.x = Ain[l16 * LDSTR + ka];
        a.y = Ain[l16 * LDSTR + ka + 1];
        // B fragment (4x16): lane = col N, VGPR0/1 = K=ka, ka+1
        b.x = wp[(size_t)ka * DTOT];
        b.y = wp[(size_t)(ka + 1) * DTOT];
        acc = __builtin_amdgcn_wmma_f32_16x16x4_f32(
            /*neg_a=*/false, a, /*neg_b=*/false, b,
            /*c_mod=*/(short)0, acc, /*reuse_a=*/false, /*reuse_b=*/false);
    }

    // Epilogue: C/D layout -> VGPR j holds rows (half*8 + j), col = ncol.
    const float bv = bias[ncol];
    #pragma unroll
    for (int j = 0; j < 8; ++j) {
        int node = row0 + half * 8 + j;
        if (node < N) {
            float o = acc[j] + bv;
            y[(size_t)node * DTOT + ncol] = fmaxf(o, 0.f);
        }
    }
}

// ---------------------------------------------------------------------------
// Host-side launch: concat -> 3x (zero agg, scatter, wmma-gemm), ping-pong
// between ws bufA and d_out. Graph-capture safe: stream-only, deterministic.
// ---------------------------------------------------------------------------
extern "C" void kernel_launch(void* const* d_in, const int* in_sizes, int n_in,
                              void* d_out, int out_size, void* d_ws, size_t ws_size,
                              hipStream_t stream) {
    const float* X     = (const float*)d_in[0];  // [N,64]
    const float* h     = (const float*)d_in[1];  // [N,32]
    const float* noise = (const float*)d_in[2];  // [N,32]
    const int*   src   = (const int*)  d_in[3];  // [E]
    const int*   dst   = (const int*)  d_in[4];  // [E]
    const float* Ws    = (const float*)d_in[5];  // [3,128,128]
    const float* bs    = (const float*)d_in[6];  // [3,128]
    const float* eps   = (const float*)d_in[7];  // [3]
    float* out = (float*)d_out;

    const int N = in_sizes[0] / 64;
    const int E = in_sizes[3];
    const int total = N * DTOT;

    float* bufA = (float*)d_ws;                 // [N,128]
    float* agg  = bufA + (size_t)total;         // [N,128]

    gin_concat<<<(total + 255) / 256, 256, 0, stream>>>(X, noise, h, bufA, N);

    const float* cur = bufA;
    for (int l = 0; l < 3; ++l) {
        gin_zero4<<<((total / 4) + 255) / 256, 256, 0, stream>>>((float4*)agg, total / 4);
        gin_scatter<<<(E + 7) / 8, 256, 0, stream>>>(cur, src, dst, agg, E);
        float* nxt = (l == 1) ? bufA : out;
        gin_gemm_relu<<<(N + 15) / 16, 256, 0, stream>>>(
            cur, agg, Ws + (size_t)l * DTOT * DTOT, bs + (size_t)l * DTOT, eps + l, nxt, N);
        cur = nxt;
    }
}